// BroadcastZInvariantPointAttention_2430951489771
// MI455X (gfx1250) — compile-verified
//
#include <hip/hip_runtime.h>
#include <hip/hip_bf16.h>
#include <math.h>

// ---------------- constants (match reference) ----------------
#define NRES   448
#define NRIG   512
#define CS     384
#define CZ     128
#define CH     16
#define NH     12
#define PQ     4
#define PV     8
#define HC     (NH*CH)          // 192
#define FEAT   960              // H*(CZ/4 + CH + PV*4)
#define INFV   100000.0f
#define EPSV   1e-8f

#define C1     0.14433756729740643f   // sqrt(1/(3*CH))
#define CB     0.57735026918962576f   // sqrt(1/3)
#define CHW    0.13608276348795434f   // sqrt(1/54)

#define SCS    516                    // padded score-row stride (bank-conflict free)
#define QS     33                     // padded Qext stride

// Async global->LDS copy path (CDNA5 GLOBAL_LOAD_ASYNC_TO_LDS_B128, ASYNCcnt).
// Probe (round 3) confirmed the builtin exists and takes int4-vector pointers:
// param0 = global (AS1) int4*, so: (global src, lds dst, imm offset, imm cpol).
#if defined(__gfx1250__) && \
    __has_builtin(__builtin_amdgcn_global_load_async_to_lds_b128) && \
    __has_builtin(__builtin_amdgcn_s_wait_asynccnt)
#define USE_ASYNC_COPY 1
typedef int vi4 __attribute__((vector_size(4 * sizeof(int))));
typedef __attribute__((address_space(1))) vi4 as1_vi4;
typedef __attribute__((address_space(3))) vi4 as3_vi4;
#else
#define USE_ASYNC_COPY 0
#endif

// ---------------- WMMA fp32 16x16x4 ----------------
typedef __attribute__((ext_vector_type(2))) float v2f;
typedef __attribute__((ext_vector_type(8))) float v8f;

__device__ __forceinline__ v8f wmma4(v2f a, v2f b, v8f c) {
  return __builtin_amdgcn_wmma_f32_16x16x4_f32(false, a, false, b, (short)0, c,
                                               false, false);
}
// Fragment maps (wave32):
//  A 16x4 : m = lane&15, kHi = (lane>>4)*2 ; a.x=A[m][k0+kHi], a.y=A[m][k0+kHi+1]
//  B 4x16 : n = lane&15 ; b.x=B[k0+kHi][n], b.y=B[k0+kHi+1][n]
//  C 16x16: vgpr r -> row = r + (lane>>4)*8, col = lane&15

// =====================================================================
// Kernel 1: fused z projections (the 103 MB roofline stream, read once).
//   b_pair16 = sqrt(1/3)*(z@w_b + b_b)  [448*448, 16]  (cols 12..15 = junk pad)
//   pair_z   = z@w_dz + b_dz            [448*448, 32]
// Block = 128 threads (4 waves) = 64 z rows. z tile staged into LDS (async
// copy when available), weights staged once per block; padded LDS strides
// keep fragment reads bank-conflict free. No conditional loads in WMMA loop.
// =====================================================================
#define ZROWS 64
__global__ __launch_bounds__(128) void k_pair_proj(
    const float* __restrict__ z,
    const float* __restrict__ w_b,  const float* __restrict__ b_b,
    const float* __restrict__ w_dz, const float* __restrict__ b_dz,
    float* __restrict__ b_pair16, float* __restrict__ pair_z)
{
  const int tid = threadIdx.x, lane = tid & 31, wv = tid >> 5;
  const int mRow = lane & 15, kHi = (lane >> 4) * 2, nCol = lane & 15;
  const size_t m0 = (size_t)blockIdx.x * ZROWS;

  __shared__ float s_wb[CZ * 17];       // w_b padded to 16 cols, stride 17
  __shared__ float s_wz[CZ * 33];       // w_dz, stride 33
  __shared__ float s_z[ZROWS * 132];    // 64 z rows, stride 132
  __shared__ float s_bb[16];

#if USE_ASYNC_COPY
  // CDNA5 async bulk copy: LDS written directly by the memory system,
  // no VGPR round-trip; tracked by ASYNCcnt.
  for (int e = tid; e < ZROWS * (CZ / 4); e += 128) {
    const int r = e >> 5, c4 = e & 31;
    __builtin_amdgcn_global_load_async_to_lds_b128(
        (as1_vi4*)(z + (m0 + r) * CZ + c4 * 4),
        (as3_vi4*)(s_z + r * 132 + c4 * 4),
        0, 0);
  }
#endif

  for (int e = tid; e < CZ * 16; e += 128) {
    const int k = e >> 4, c = e & 15;
    s_wb[k * 17 + c] = (c < NH) ? w_b[k * NH + c] : 0.0f;
  }
  for (int e = tid; e < CZ * 32; e += 128) {
    const int k = e >> 5, c = e & 31;
    s_wz[k * 33 + c] = w_dz[e];
  }
  if (tid < 16) s_bb[tid] = (tid < NH) ? b_b[tid] : 0.0f;

#if USE_ASYNC_COPY
  __builtin_amdgcn_s_wait_asynccnt(0);
#else
  for (int e = tid; e < ZROWS * (CZ / 4); e += 128) {   // coalesced b128 stream
    const int r = e >> 5, c4 = e & 31;
    const float4 vz = *(const float4*)(z + (m0 + r) * CZ + c4 * 4);
    *(float4*)(s_z + r * 132 + c4 * 4) = vz;
  }
#endif
  __syncthreads();

  const float* zrow = s_z + (wv * 16 + mRow) * 132;
  v8f accB = {}, accZ0 = {}, accZ1 = {};
  for (int k0 = 0; k0 < CZ; k0 += 4) {
    const int k1 = k0 + kHi, k2 = k1 + 1;
    v2f a;  a.x  = zrow[k1];               a.y  = zrow[k2];
    v2f bB; bB.x = s_wb[k1 * 17 + nCol];   bB.y = s_wb[k2 * 17 + nCol];
    accB = wmma4(a, bB, accB);
    v2f b0; b0.x = s_wz[k1 * 33 + nCol];   b0.y = s_wz[k2 * 33 + nCol];
    accZ0 = wmma4(a, b0, accZ0);
    v2f b1; b1.x = s_wz[k1 * 33 + 16 + nCol]; b1.y = s_wz[k2 * 33 + 16 + nCol];
    accZ1 = wmma4(a, b1, accZ1);
  }
#pragma unroll
  for (int r = 0; r < 8; ++r) {
    const int row = r + (lane >> 4) * 8;
    const size_t m = m0 + wv * 16 + row;
    b_pair16[m * 16 + nCol]   = CB * (accB[r] + s_bb[nCol]);
    pair_z[m * 32 + nCol]      = accZ0[r] + b_dz[nCol];
    pair_z[m * 32 + 16 + nCol] = accZ1[r] + b_dz[16 + nCol];
  }
}

// =====================================================================
// Kernel 2: s projections, 512x384 @ 384x{192,384,144,432}
// =====================================================================
__global__ __launch_bounds__(32) void k_s_proj(
    const float* __restrict__ s,
    const float* __restrict__ w_q,   const float* __restrict__ b_q,
    const float* __restrict__ w_kv,  const float* __restrict__ b_kv,
    const float* __restrict__ w_qp,  const float* __restrict__ b_qp,
    const float* __restrict__ w_kvp, const float* __restrict__ b_kvp,
    float* __restrict__ q_lin, float* __restrict__ kv_lin,
    float* __restrict__ qp_lin, float* __restrict__ kvp_lin)
{
  const int lane = threadIdx.x & 31;
  const int mRow = lane & 15, kHi = (lane >> 4) * 2, nCol = lane & 15;
  const int m0 = blockIdx.x * 16;

  int t = blockIdx.y;
  const float* W; const float* bias; float* out; int N;
  if (t < 12)              { W = w_q;   bias = b_q;   out = q_lin;   N = 192; }
  else if ((t -= 12) < 24) { W = w_kv;  bias = b_kv;  out = kv_lin;  N = 384; }
  else if ((t -= 24) < 9)  { W = w_qp;  bias = b_qp;  out = qp_lin;  N = 144; }
  else { t -= 9;             W = w_kvp; bias = b_kvp; out = kvp_lin; N = 432; }
  const int n0 = t * 16;

  const float* arow = s + (size_t)(m0 + mRow) * CS;
  v8f acc = {};
  for (int k0 = 0; k0 < CS; k0 += 4) {
    v2f a; a.x = arow[k0 + kHi]; a.y = arow[k0 + kHi + 1];
    v2f b; b.x = W[(size_t)(k0 + kHi) * N + n0 + nCol];
           b.y = W[(size_t)(k0 + kHi + 1) * N + n0 + nCol];
    acc = wmma4(a, b, acc);
  }
  const float bb = bias[n0 + nCol];
#pragma unroll
  for (int r = 0; r < 8; ++r) {
    const int row = r + (lane >> 4) * 8;
    out[(size_t)(m0 + row) * N + n0 + nCol] = acc[r] + bb;
  }
}

// =====================================================================
// Kernel 3: rigid transform of points + per-(i,h) squared norms
// =====================================================================
__global__ void k_points(
    const float* __restrict__ qp_lin, const float* __restrict__ kvp_lin,
    const float* __restrict__ rot, const float* __restrict__ trans,
    float* __restrict__ q_pts, float* __restrict__ k_pts, float* __restrict__ v_pts,
    float* __restrict__ qn2, float* __restrict__ kn2)
{
  const int i = blockIdx.x;
  const int t = threadIdx.x;
  __shared__ float R[9], T[3];
  __shared__ float s_qp[48 * 3], s_kp[48 * 3];
  if (t < 9) R[t] = rot[i * 9 + t];
  if (t < 3) T[t] = trans[i * 3 + t];
  __syncthreads();

  for (int d = t; d < 48; d += blockDim.x) {
    const float lx = qp_lin[i * 144 + d];
    const float ly = qp_lin[i * 144 + 48 + d];
    const float lz = qp_lin[i * 144 + 96 + d];
    const float gx = R[0]*lx + R[1]*ly + R[2]*lz + T[0];
    const float gy = R[3]*lx + R[4]*ly + R[5]*lz + T[1];
    const float gz = R[6]*lx + R[7]*ly + R[8]*lz + T[2];
    float* o = q_pts + ((size_t)i * 48 + d) * 3;
    o[0] = gx; o[1] = gy; o[2] = gz;
    s_qp[d*3+0] = gx; s_qp[d*3+1] = gy; s_qp[d*3+2] = gz;
  }
  for (int d = t; d < 144; d += blockDim.x) {
    const float lx = kvp_lin[i * 432 + d];
    const float ly = kvp_lin[i * 432 + 144 + d];
    const float lz = kvp_lin[i * 432 + 288 + d];
    const float gx = R[0]*lx + R[1]*ly + R[2]*lz + T[0];
    const float gy = R[3]*lx + R[4]*ly + R[5]*lz + T[1];
    const float gz = R[6]*lx + R[7]*ly + R[8]*lz + T[2];
    const int h = d / 12, pp = d % 12;
    if (pp < PQ) {
      const int kd = h * PQ + pp;
      float* o = k_pts + ((size_t)i * 48 + kd) * 3;
      o[0] = gx; o[1] = gy; o[2] = gz;
      s_kp[kd*3+0] = gx; s_kp[kd*3+1] = gy; s_kp[kd*3+2] = gz;
    } else {
      float* o = v_pts + ((size_t)i * 96 + h * PV + (pp - PQ)) * 3;
      o[0] = gx; o[1] = gy; o[2] = gz;
    }
  }
  __syncthreads();
  if (t < NH) {
    float sq = 0.f, sk = 0.f;
    for (int p = 0; p < PQ; ++p)
      for (int c = 0; c < 3; ++c) {
        const float a = s_qp[(t*PQ+p)*3+c]; sq += a * a;
        const float b = s_kp[(t*PQ+p)*3+c]; sk += b * b;
      }
    qn2[i * NH + t] = sq;
    kn2[i * NH + t] = sk;
  }
}

// =====================================================================
// Kernel 4: build padded, K-major fragment sources so attention WMMA
// loops have zero conditional loads and coalesced B-fragments:
//   kextT[h][k(32)][j(512)] : k<16 -> k-proj, k<28 -> k_pts, else 0
//   vext [h][j(512)][c(48)] : c<16 -> v-proj, c<40 -> v_pts,  else 0
// =====================================================================
__global__ void k_prep_ext(const float* __restrict__ kv_lin,
                           const float* __restrict__ k_pts,
                           const float* __restrict__ v_pts,
                           float* __restrict__ kextT, float* __restrict__ vext)
{
  const int idx = blockIdx.x * blockDim.x + threadIdx.x;
  const int totK = NH * 32 * NRIG;       // 196608
  if (idx < totK) {
    const int h = idx / (32 * NRIG);
    const int k = (idx / NRIG) & 31;
    const int j = idx & (NRIG - 1);
    float v = 0.0f;
    if (k < 16)      v = kv_lin[(size_t)j * 384 + h * 32 + k];
    else if (k < 28) v = k_pts[((size_t)j * 48 + h * PQ) * 3 + (k - 16)];
    kextT[idx] = v;
  }
  const int totV = NH * NRIG * 48;       // 294912
  if (idx < totV) {
    const int h = idx / (NRIG * 48);
    const int j = (idx / 48) & (NRIG - 1);
    const int c = idx % 48;
    float v = 0.0f;
    if (c < 16)      v = kv_lin[(size_t)j * 384 + h * 32 + 16 + c];
    else if (c < 40) v = v_pts[((size_t)j * 96 + h * PV) * 3 + (c - 16)];
    vext[idx] = v;
  }
}

// =====================================================================
// Kernel 5: attention core. Block = 128 threads (4 waves) per
// (16 query rigids, head).  ||q-k||^2 expansion folds the point term
// into one K=28(pad 32) WMMA GEMM.  Wave-uniform branches via
// readfirstlane so EXEC is never masked around WMMA.
// =====================================================================
__global__ __launch_bounds__(128) void k_attn(
    const float* __restrict__ q_lin, const float* __restrict__ kextT,
    const float* __restrict__ vext, const float* __restrict__ q_pts,
    const float* __restrict__ qn2, const float* __restrict__ kn2,
    const float* __restrict__ b_pair16,
    const float* __restrict__ mask, const int* __restrict__ ridx,
    const float* __restrict__ hwts,
    float* __restrict__ feats, float* __restrict__ o_pt_g,
    float* __restrict__ a_agg)
{
  const int tid = threadIdx.x, lane = tid & 31;
  const int wvu = __builtin_amdgcn_readfirstlane(tid >> 5);
  const int mRow = lane & 15, kHi = (lane >> 4) * 2, nCol = lane & 15;
  const int h = blockIdx.y, i0 = blockIdx.x * 16;

  __shared__ float s_q[16 * QS];        // Qext, padded stride
  __shared__ float s_sc[16 * SCS];      // scores/probs, padded stride
  __shared__ float s_agg[16 * NRES];    // per-residue aggregation
  __shared__ float s_red[16 * 8];
  __shared__ int   s_ri[16];

  const float hw = logf(1.0f + expf(hwts[h])) * CHW;

  for (int e = tid; e < 16 * 32; e += 128) {
    const int r = e >> 5, c = e & 31;
    const int i = i0 + r;
    float v = 0.0f;
    if (c < 16)      v = q_lin[(size_t)i * HC + h * CH + c] * C1;
    else if (c < 28) v = q_pts[((size_t)i * 48 + h * PQ) * 3 + (c - 16)] * hw;
    s_q[r * QS + c] = v;
  }
  for (int e = tid; e < 16 * NRES; e += 128) s_agg[e] = 0.0f;
  if (tid < 16) s_ri[tid] = ridx[i0 + tid];
  __syncthreads();

  float rowQn[8], rowMi[8];
#pragma unroll
  for (int r = 0; r < 8; ++r) {
    const int i = i0 + r + (lane >> 4) * 8;
    rowQn[r] = qn2[i * NH + h];
    rowMi[r] = mask[i];
  }

  // ---- pass 1: scores (each wave: 8 j-tiles) ----
  const float* kT = kextT + (size_t)h * 32 * NRIG;
  for (int jt = wvu * 8; jt < wvu * 8 + 8; ++jt) {
    const int j0 = jt * 16, j = j0 + nCol;
    v8f acc = {};
#pragma unroll
    for (int k0 = 0; k0 < 32; k0 += 4) {
      v2f a; a.x = s_q[mRow * QS + k0 + kHi]; a.y = s_q[mRow * QS + k0 + kHi + 1];
      v2f b; b.x = kT[(size_t)(k0 + kHi) * NRIG + j];
             b.y = kT[(size_t)(k0 + kHi + 1) * NRIG + j];
      acc = wmma4(a, b, acc);
    }
    const int   rj = ridx[j];
    const float mj = mask[j];
    const float kn = kn2[j * NH + h];
#pragma unroll
    for (int r = 0; r < 8; ++r) {
      const int row = r + (lane >> 4) * 8;
      float sc = acc[r];
      sc += -0.5f * hw * (rowQn[r] + kn);
      sc += b_pair16[((size_t)s_ri[row] * NRES + rj) * 16 + h];
      sc += INFV * (rowMi[r] * mj - 1.0f);
      s_sc[row * SCS + j] = sc;
    }
  }
  __syncthreads();

  // ---- softmax + residue aggregation (128 threads: row = tid>>3, 64 j each)
  {
    const int row = tid >> 3, seg = tid & 7;
    float* rp = s_sc + row * SCS + seg * 64;
    float mx = -3.0e38f;
    for (int u = 0; u < 64; ++u) mx = fmaxf(mx, rp[u]);
    s_red[row * 8 + seg] = mx;
    __syncthreads();
    float rmax = s_red[row * 8];
#pragma unroll
    for (int u = 1; u < 8; ++u) rmax = fmaxf(rmax, s_red[row * 8 + u]);
    float sum = 0.0f;
    for (int u = 0; u < 64; ++u) { const float e = __expf(rp[u] - rmax); rp[u] = e; sum += e; }
    __syncthreads();
    s_red[row * 8 + seg] = sum;
    __syncthreads();
    float rsum = 0.0f;
#pragma unroll
    for (int u = 0; u < 8; ++u) rsum += s_red[row * 8 + u];
    const float inv = 1.0f / rsum;
    const int jb = seg * 64;
    for (int u = 0; u < 64; ++u) {
      const float a = rp[u] * inv;
      rp[u] = a;
      atomicAdd(&s_agg[row * NRES + ridx[jb + u]], a);   // ds_add_f32
    }
  }
  __syncthreads();

  // ---- pass 2: O = A.V, O_pt = A.v_pts; one N-tile per wave ----
  if (wvu < 3) {
    const int c0 = wvu * 16;
    const float* vx = vext + (size_t)h * NRIG * 48;
    v8f acc = {};
    for (int jt = 0; jt < 32; ++jt) {
#pragma unroll
      for (int ks = 0; ks < 16; ks += 4) {
        const int kb = jt * 16 + ks;
        v2f a; a.x = s_sc[mRow * SCS + kb + kHi];
               a.y = s_sc[mRow * SCS + kb + kHi + 1];
        const int j1 = kb + kHi, j2 = j1 + 1;
        v2f b; b.x = vx[(size_t)j1 * 48 + c0 + nCol];
               b.y = vx[(size_t)j2 * 48 + c0 + nCol];
        acc = wmma4(a, b, acc);
      }
    }
#pragma unroll
    for (int r = 0; r < 8; ++r) {
      const int row = r + (lane >> 4) * 8;
      const int i = i0 + row;
      if (wvu == 0)      feats[(size_t)i * FEAT + h * CH + nCol] = acc[r];
      else if (wvu == 1) o_pt_g[((size_t)i * NH + h) * 24 + nCol] = acc[r];
      else if (nCol < 8) o_pt_g[((size_t)i * NH + h) * 24 + 16 + nCol] = acc[r];
    }
  }
  // flush a_agg (layout [h][i][n]); s_agg final since the last barrier
  for (int e = tid; e < 16 * NRES; e += 128) {
    const int r = e / NRES, n = e - r * NRES;
    a_agg[((size_t)h * NRIG + i0 + r) * NRES + n] = s_agg[e];
  }
}

// =====================================================================
// Kernel 6: o_pair[i,h,d] = sum_n a_agg[h,i,n] * pair_z[ridx[i],n,d]
// M = 16 heads (12 real + 4 zero-padded rows), K = 448, N = 32.
// =====================================================================
__global__ __launch_bounds__(32) void k_opair(
    const float* __restrict__ a_agg, const float* __restrict__ pair_z,
    const int* __restrict__ ridx, float* __restrict__ feats)
{
  const int lane = threadIdx.x & 31;
  const int mRow = lane & 15, kHi = (lane >> 4) * 2, nCol = lane & 15;
  const int i = blockIdx.x;
  const int ri = ridx[i];
  const float* pz = pair_z + (size_t)ri * NRES * 32;

  v8f acc0 = {}, acc1 = {};
  for (int k0 = 0; k0 < NRES; k0 += 4) {
    const int k1 = k0 + kHi, k2 = k1 + 1;
    v2f a;
    a.x = a_agg[((size_t)mRow * NRIG + i) * NRES + k1];
    a.y = a_agg[((size_t)mRow * NRIG + i) * NRES + k2];
    v2f b0; b0.x = pz[(size_t)k1 * 32 + nCol];      b0.y = pz[(size_t)k2 * 32 + nCol];
    acc0 = wmma4(a, b0, acc0);
    v2f b1; b1.x = pz[(size_t)k1 * 32 + 16 + nCol]; b1.y = pz[(size_t)k2 * 32 + 16 + nCol];
    acc1 = wmma4(a, b1, acc1);
  }
#pragma unroll
  for (int r = 0; r < 8; ++r) {
    const int row = r + (lane >> 4) * 8;    // head index
    if (row < NH) {
      feats[(size_t)i * FEAT + 576 + row * 32 + nCol]      = acc0[r];
      feats[(size_t)i * FEAT + 576 + row * 32 + 16 + nCol] = acc1[r];
    }
  }
}

// =====================================================================
// Kernel 7: o_pt = R^T (o_pt_g - t); x/y/z/norm feature slabs
// =====================================================================
__global__ void k_finpt(const float* __restrict__ o_pt_g,
                        const float* __restrict__ rot, const float* __restrict__ trans,
                        float* __restrict__ feats)
{
  const int i = blockIdx.x;
  const int t = threadIdx.x;
  __shared__ float R[9], T[3];
  if (t < 9) R[t] = rot[i * 9 + t];
  if (t < 3) T[t] = trans[i * 3 + t];
  __syncthreads();
  if (t < NH * PV) {
    const int h = t >> 3, p = t & 7;
    const float* g = o_pt_g + ((size_t)i * NH + h) * 24 + p * 3;
    const float dx = g[0] - T[0], dy = g[1] - T[1], dz = g[2] - T[2];
    const float ox = R[0]*dx + R[3]*dy + R[6]*dz;
    const float oy = R[1]*dx + R[4]*dy + R[7]*dz;
    const float oz = R[2]*dx + R[5]*dy + R[8]*dz;
    const float nrm = sqrtf(ox*ox + oy*oy + oz*oz + EPSV);
    const int hp = h * PV + p;
    float* f = feats + (size_t)i * FEAT;
    f[192 + hp] = ox;
    f[288 + hp] = oy;
    f[384 + hp] = oz;
    f[480 + hp] = nrm;
  }
}

// =====================================================================
// Kernel 8: out = feats[512,960] @ w_out[960,384] + b_out
// =====================================================================
__global__ __launch_bounds__(32) void k_out(
    const float* __restrict__ feats, const float* __restrict__ w_out,
    const float* __restrict__ b_out, float* __restrict__ out)
{
  const int lane = threadIdx.x & 31;
  const int mRow = lane & 15, kHi = (lane >> 4) * 2, nCol = lane & 15;
  const int m0 = blockIdx.x * 16;
  const int n0 = blockIdx.y * 16;

  const float* arow = feats + (size_t)(m0 + mRow) * FEAT;
  v8f acc = {};
  for (int k0 = 0; k0 < FEAT; k0 += 4) {
    v2f a; a.x = arow[k0 + kHi]; a.y = arow[k0 + kHi + 1];
    v2f b; b.x = w_out[(size_t)(k0 + kHi) * CS + n0 + nCol];
           b.y = w_out[(size_t)(k0 + kHi + 1) * CS + n0 + nCol];
    acc = wmma4(a, b, acc);
  }
  const float bb = b_out[n0 + nCol];
#pragma unroll
  for (int r = 0; r < 8; ++r) {
    const int row = r + (lane >> 4) * 8;
    out[(size_t)(m0 + row) * CS + n0 + nCol] = acc[r] + bb;
  }
}

// =====================================================================
extern "C" void kernel_launch(void* const* d_in, const int* in_sizes, int n_in,
                              void* d_out, int out_size, void* d_ws, size_t ws_size,
                              hipStream_t stream) {
  const float* s      = (const float*)d_in[0];
  const float* z      = (const float*)d_in[1];
  const float* rot    = (const float*)d_in[2];
  const float* trans  = (const float*)d_in[3];
  const float* mask   = (const float*)d_in[4];
  const int*   ridx   = (const int*)  d_in[5];
  const float* w_q    = (const float*)d_in[6];
  const float* b_q    = (const float*)d_in[7];
  const float* w_kv   = (const float*)d_in[8];
  const float* b_kv   = (const float*)d_in[9];
  const float* w_qp   = (const float*)d_in[10];
  const float* b_qp   = (const float*)d_in[11];
  const float* w_kvp  = (const float*)d_in[12];
  const float* b_kvp  = (const float*)d_in[13];
  const float* w_b    = (const float*)d_in[14];
  const float* b_b    = (const float*)d_in[15];
  const float* w_dz   = (const float*)d_in[16];
  const float* b_dz   = (const float*)d_in[17];
  const float* hwts   = (const float*)d_in[18];
  const float* w_out  = (const float*)d_in[19];
  const float* b_out  = (const float*)d_in[20];
  float* out = (float*)d_out;

  // workspace carve-up (floats); total ~61.3 MB
  float* w = (float*)d_ws;
  float* b_pair16 = w; w += (size_t)NRES * NRES * 16;   // padded pair bias
  float* pair_z   = w; w += (size_t)NRES * NRES * 32;
  float* q_lin    = w; w += (size_t)NRIG * HC;
  float* kv_lin   = w; w += (size_t)NRIG * 2 * HC;
  float* qp_lin   = w; w += (size_t)NRIG * 144;
  float* kvp_lin  = w; w += (size_t)NRIG * 432;
  float* q_pts    = w; w += (size_t)NRIG * 144;
  float* k_pts    = w; w += (size_t)NRIG * 144;
  float* v_pts    = w; w += (size_t)NRIG * 288;
  float* qn2      = w; w += (size_t)NRIG * NH;
  float* kn2      = w; w += (size_t)NRIG * NH;
  float* o_pt_g   = w; w += (size_t)NRIG * NH * 24;
  float* a_agg    = w; w += (size_t)16 * NRIG * NRES;   // 16 heads (4 zero pad)
  float* kextT    = w; w += (size_t)NH * 32 * NRIG;
  float* vext     = w; w += (size_t)NH * NRIG * 48;
  float* feats    = w; w += (size_t)NRIG * FEAT;

  // zero the 4 padded head rows of a_agg once per call (graph-capture safe)
  (void)hipMemsetAsync(a_agg + (size_t)NH * NRIG * NRES, 0,
                       (size_t)(16 - NH) * NRIG * NRES * sizeof(float), stream);

  // 1) fused z projections (single pass over the 103 MB z tensor)
  k_pair_proj<<<(NRES * NRES) / ZROWS, 128, 0, stream>>>(z, w_b, b_b, w_dz, b_dz,
                                                         b_pair16, pair_z);
  // 2) s projections
  k_s_proj<<<dim3(NRIG / 16, 72), 32, 0, stream>>>(s, w_q, b_q, w_kv, b_kv,
                                                   w_qp, b_qp, w_kvp, b_kvp,
                                                   q_lin, kv_lin, qp_lin, kvp_lin);
  // 3) rigid transforms + norms
  k_points<<<NRIG, 64, 0, stream>>>(qp_lin, kvp_lin, rot, trans,
                                    q_pts, k_pts, v_pts, qn2, kn2);
  // 4) padded K-major fragment staging
  k_prep_ext<<<(NH * NRIG * 48 + 255) / 256, 256, 0, stream>>>(kv_lin, k_pts, v_pts,
                                                               kextT, vext);
  // 5) attention core
  k_attn<<<dim3(NRIG / 16, NH), 128, 0, stream>>>(q_lin, kextT, vext, q_pts,
                                                  qn2, kn2, b_pair16, mask, ridx,
                                                  hwts, feats, o_pt_g, a_agg);
  // 6) pair aggregation output
  k_opair<<<NRIG, 32, 0, stream>>>(a_agg, pair_z, ridx, feats);
  // 7) inverse point transform + norms
  k_finpt<<<NRIG, 96, 0, stream>>>(o_pt_g, rot, trans, feats);
  // 8) final projection
  k_out<<<dim3(NRIG / 16, CS / 16), 32, 0, stream>>>(feats, w_out, b_out, out);
}